// TopKTopPSampler_89678917141260
// MI455X (gfx1250) — compile-verified
//
#include <hip/hip_runtime.h>
#include <stdint.h>

constexpr int kB     = 128;
constexpr int kV     = 128000;
constexpr int NT     = 256;          // 8 wave32s
constexpr int NBINS  = 4096;         // 12-bit radix level
constexpr int CAND   = 2048;         // k <= 1023 + boundary-bin slack
constexpr int CHUNK  = NT * 4;       // 1024 floats per streamed chunk
constexpr int NCHUNK = kV / CHUNK;   // 125

// Monotonic float -> uint key (order-preserving for all finite floats).
__device__ __forceinline__ uint32_t fkey(float f) {
    uint32_t u = __float_as_uint(f);
    return u ^ (uint32_t(int32_t(u) >> 31) | 0x80000000u);
}

template <int N>
__device__ __forceinline__ void wait_asynccnt() {
    asm volatile("s_wait_asynccnt %0" :: "i"(N) : "memory");
}
__device__ __forceinline__ void wait_dscnt0() {
    asm volatile("s_wait_dscnt 0" ::: "memory");
}
// CDNA5 async global->LDS copy, 16B per lane. VDST = LDS byte offset VGPR,
// VADDR = 64-bit global address, no SADDR ("off").
__device__ __forceinline__ void async_load_b128_to_lds(const void* gaddr, uint32_t lds_off) {
    asm volatile("global_load_async_to_lds_b128 %0, %1, off"
                 :: "v"(lds_off), "v"(gaddr)
                 : "memory");
}

// Stream one 128000-float row through LDS with double-buffered async copies.
// Each lane consumes only the 16B it loaded, so only per-wave waits are needed.
template <typename F>
__device__ __forceinline__ void stream_row(const float* __restrict__ L,
                                           int tid, float4* stage4, F f) {
    const uint32_t lds_off0 = (uint32_t)(uintptr_t)(&stage4[tid]);
    const uint32_t lds_off1 = (uint32_t)(uintptr_t)(&stage4[NT + tid]);
    async_load_b128_to_lds(L + tid * 4, lds_off0);                 // chunk 0
    for (int c = 0; c < NCHUNK; ++c) {
        const int buf = c & 1;
        if (c + 1 < NCHUNK) {
            async_load_b128_to_lds(L + (size_t)(c + 1) * CHUNK + tid * 4,
                                   buf ? lds_off0 : lds_off1);
            wait_asynccnt<1>();       // chunk c landed (in-order completion)
        } else {
            wait_asynccnt<0>();
        }
        if (c + 4 < NCHUNK)           // deep prefetch (global_prefetch_b8)
            __builtin_prefetch(L + (size_t)(c + 4) * CHUNK + tid * 4, 0, 0);
        float4 v = stage4[buf * NT + tid];
        wait_dscnt0();                // regs hold data before buffer is reused
        const int i0 = c * CHUNK + tid * 4;
        f(v.x, i0); f(v.y, i0 + 1); f(v.z, i0 + 2); f(v.w, i0 + 3);
    }
}

__global__ __launch_bounds__(NT)
void topk_topp_sample_kernel(const float* __restrict__ logits,
                             const int*   __restrict__ kk,
                             const float* __restrict__ pp,
                             const float* __restrict__ qq,
                             int*         __restrict__ out) {
    __shared__ uint32_t hist[NBINS];
    __shared__ float    skey[CAND];
    __shared__ int      sidx[CAND];
    __shared__ float    wa[CAND];
    __shared__ float    wb[CAND];
    __shared__ float4   stage4[2 * NT];          // async staging (8 KB)
    __shared__ uint32_t s_cnt, s_bin, s_rank, s_kth;

    const int row = blockIdx.x;
    const int tid = threadIdx.x;
    const float* __restrict__ L = logits + (size_t)row * kV;
    const int   k = kk[row];
    const float p = pp[row];

    // ---- Pass A (HBM): level-0 histogram on key bits [31:20] ---------------
    for (int i = tid; i < NBINS; i += NT) hist[i] = 0;
    __syncthreads();
    stream_row(L, tid, stage4, [&](float val, int) {
        atomicAdd(&hist[fkey(val) >> 20], 1u);
    });
    __syncthreads();
    if (tid == 0) {   // locate the bin holding the k-th largest (from the top)
        uint32_t need = (uint32_t)k, cum = 0;
        int b = NBINS - 1;
        for (; b > 0; --b) {
            uint32_t c = hist[b];
            if (cum + c >= need) break;
            cum += c;
        }
        s_bin = (uint32_t)b; s_rank = need - cum;
    }
    __syncthreads();
    const uint32_t b0 = s_bin;
    const uint32_t r1 = s_rank;

    // ---- Pass B (L2): level-1 histogram on key bits [19:8] -----------------
    for (int i = tid; i < NBINS; i += NT) hist[i] = 0;
    __syncthreads();
    stream_row(L, tid, stage4, [&](float val, int) {
        const uint32_t key = fkey(val);
        if ((key >> 20) == b0) atomicAdd(&hist[(key >> 8) & 0xFFFu], 1u);
    });
    __syncthreads();
    if (tid == 0) {
        uint32_t need = r1, cum = 0;
        int b = NBINS - 1;
        for (; b > 0; --b) {
            uint32_t c = hist[b];
            if (cum + c >= need) break;
            cum += c;
        }
        s_bin = (uint32_t)b; s_rank = need - cum;
    }
    __syncthreads();
    const uint32_t b1   = s_bin;
    const uint32_t r2   = s_rank;
    const uint32_t pref = (b0 << 12) | b1;       // key >> 8 of the k-th largest

    // ---- Pass C (L2, fused): low-byte histogram of the boundary bin AND ----
    // ---- gather of every element with key>>8 >= pref (superset of top-k) ---
    for (int i = tid; i < 256; i += NT) hist[i] = 0;
    if (tid == 0) s_cnt = 0;
    __syncthreads();
    stream_row(L, tid, stage4, [&](float val, int idx) {
        const uint32_t key = fkey(val);
        const uint32_t hi  = key >> 8;
        if (hi == pref) atomicAdd(&hist[key & 0xFFu], 1u);
        if (hi >= pref) {
            const uint32_t pos = atomicAdd(&s_cnt, 1u);
            if (pos < CAND) { skey[pos] = val; sidx[pos] = idx; }
        }
    });
    __syncthreads();
    const int n = min((int)s_cnt, CAND);
    if (tid == 0) {
        uint32_t need = r2, cum = 0;
        int b = 255;
        for (; b > 0; --b) {
            uint32_t c = hist[b];
            if (cum + c >= need) break;
            cum += c;
        }
        s_kth = (pref << 8) | (uint32_t)b;        // exact key of k-th largest
    }
    __syncthreads();
    const uint32_t kth = s_kth;

    // Apply exact top-k mask to over-collected boundary elements (w -> 0).
    for (int i = tid; i < n; i += NT)
        if (fkey(skey[i]) < kth) skey[i] = -__builtin_inff();
    // Pad the rest for the power-of-two sort.
    for (int i = tid; i < CAND; i += NT)
        if (i >= n) { skey[i] = -__builtin_inff(); sidx[i] = 0x7FFFFFFF; }
    __syncthreads();

    // Bitonic sort, descending by logit, N = 2048, 256 threads.
    for (int k2 = 2; k2 <= CAND; k2 <<= 1) {
        for (int j = k2 >> 1; j > 0; j >>= 1) {
            for (int i = tid; i < CAND; i += NT) {
                const int ixj = i ^ j;
                if (ixj > i) {
                    const bool dir = ((i & k2) == 0);
                    const float a = skey[i], b = skey[ixj];
                    if ((a < b) == dir) {
                        skey[i] = b; skey[ixj] = a;
                        const int t = sidx[i]; sidx[i] = sidx[ixj]; sidx[ixj] = t;
                    }
                }
            }
            __syncthreads();
        }
    }

    // Softmax weights + inclusive scan (Hillis-Steele, ping-pong wa/wb).
    const float m = skey[0];                       // row max (rank 0)
    for (int i = tid; i < CAND; i += NT)
        wa[i] = (i < n) ? __expf(skey[i] - m) : 0.0f;
    __syncthreads();
    float* src = wa; float* dst = wb;
    for (int off = 1; off < CAND; off <<= 1) {
        for (int i = tid; i < CAND; i += NT)
            dst[i] = src[i] + ((i >= off) ? src[i - off] : 0.0f);
        __syncthreads();
        float* t = src; src = dst; dst = t;
    }
    const float Z  = src[CAND - 1];
    const float pZ = p * Z;

    // Nucleus cut: keep while exclusive descending prefix < p*Z; then
    // argmax of w/q over survivors == reference argmax(probs/q).
    float bestv = -1.0f; uint32_t besti = 0xFFFFFFFFu;
    for (int i = tid; i < CAND; i += NT) {
        if (i < n) {
            const float w = __expf(skey[i] - m);
            const float E = src[i] - w;            // exclusive prefix
            if (w > 0.0f && E < pZ) {
                const int vi = sidx[i];
                const float val = w / qq[(size_t)row * kV + vi];
                if (val > bestv || (val == bestv && (uint32_t)vi < besti)) {
                    bestv = val; besti = (uint32_t)vi;
                }
            }
        }
    }
    __syncthreads();
    wb[tid] = bestv; hist[tid] = besti;            // reuse LDS for reduction
    __syncthreads();
    for (int s = NT / 2; s > 0; s >>= 1) {
        if (tid < s) {
            const float ov = wb[tid + s]; const uint32_t oi = hist[tid + s];
            if (ov > wb[tid] || (ov == wb[tid] && oi < hist[tid])) {
                wb[tid] = ov; hist[tid] = oi;
            }
        }
        __syncthreads();
    }
    if (tid == 0) out[row] = (int)hist[0];
}

extern "C" void kernel_launch(void* const* d_in, const int* in_sizes, int n_in,
                              void* d_out, int out_size, void* d_ws, size_t ws_size,
                              hipStream_t stream) {
    const float* logits = (const float*)d_in[0];
    const int*   k      = (const int*)d_in[1];
    const float* p      = (const float*)d_in[2];
    const float* q      = (const float*)d_in[3];
    int* out = (int*)d_out;
    (void)in_sizes; (void)n_in; (void)out_size; (void)d_ws; (void)ws_size;
    topk_topp_sample_kernel<<<kB, NT, 0, stream>>>(logits, k, p, q, out);
}